// ProbsApproxCatMultiLayer_70995809402947
// MI455X (gfx1250) — compile-verified
//
#include <hip/hip_runtime.h>
#include <hip/hip_bf16.h>

#define MUX_IN   32
#define MUX_OUT  8
#define EPSF     1e-20f

#define TPB      256     // 8 wave32 per block
#define ITERS    32      // float4 iterations per thread
#define DEPTH    4       // async pipeline depth (power of 2)

typedef float f4 __attribute__((ext_vector_type(4)));

// ---------------------------------------------------------------------------
// Kernel 1: per-batch Gumbel top-8 multi-hot mask (forward value of the
// straight-through estimator). One wave32 per batch.
// ---------------------------------------------------------------------------
__global__ void choice_kernel(const float* __restrict__ u,
                              const float* __restrict__ logits,
                              float* __restrict__ choice) {
    const int b = blockIdx.x;
    const int c = threadIdx.x;                       // 0..31 == MUX_IN lanes
    const float uv = u[b * MUX_IN + c];
    const float lg = logits[c];
    const float gn = -__logf(-__logf(uv + EPSF) + EPSF);   // Gumbel noise
    const float p  = lg + gn;                        // perturbed logit

    // rank among 32 lanes; tie-break: lower index wins (matches lax.top_k)
    int rank = 0;
#pragma unroll
    for (int j = 0; j < MUX_IN; ++j) {
        const float pj = __shfl(p, j, MUX_IN);
        rank += (pj > p) || ((pj == p) && (j < c));
    }
    choice[b * MUX_IN + c] = (rank < MUX_OUT) ? 1.0f : 0.0f;
}

// ---------------------------------------------------------------------------
// s_wait_asynccnt helper (immediate-only SOPP; async loads complete in order)
// ---------------------------------------------------------------------------
__device__ __forceinline__ void wait_async(int n) {
    switch (n) {
    case 0:  asm volatile("s_wait_asynccnt 0x0" ::: "memory"); break;
    case 1:  asm volatile("s_wait_asynccnt 0x1" ::: "memory"); break;
    case 2:  asm volatile("s_wait_asynccnt 0x2" ::: "memory"); break;
    default: asm volatile("s_wait_asynccnt 0x3" ::: "memory"); break;
    }
}

__device__ __forceinline__ void async_load_b128_nt(unsigned lds_off,
                                                   unsigned glb_off,
                                                   const float* sbase) {
    // GVS mode: copies global saddr+vaddr into LDS at vdst; tracked by ASYNCcnt.
    // TH_LOAD_NT: read-once 256MB stream must not thrash the 192MB L2.
    asm volatile("global_load_async_to_lds_b128 %0, %1, %2 th:TH_LOAD_NT"
                 :
                 : "v"(lds_off), "v"(glb_off), "s"(sbase)
                 : "memory");
}

// ---------------------------------------------------------------------------
// Kernel 2: out[b,i,c] = in[b,i,c] * choice[b,c]
// Bandwidth-bound streaming multiply (0.125 FLOP/byte; ~22us at 23.3 TB/s).
// 4-deep async-to-LDS pipeline, non-temporal B128 loads and stores.
// Each thread owns its private LDS slot: no barriers required (ASYNCcnt is
// per-wave, every wave consumes only what it issued).
// ---------------------------------------------------------------------------
__global__ void scale_kernel(const float* __restrict__ in,
                             const float* __restrict__ choice,
                             float* __restrict__ out,
                             int chunks_per_batch) {
    __shared__ __align__(16) f4 lbuf[DEPTH][TPB];

    const int tid   = threadIdx.x;
    const int b     = blockIdx.x / chunks_per_batch;
    const int chunk = blockIdx.x % chunks_per_batch;

    const size_t f4_per_block = (size_t)TPB * ITERS;
    const size_t base_f4 = ((size_t)b * chunks_per_batch + chunk) * f4_per_block;

    // per-thread channel phase is constant: (f4 index) mod 8 == tid mod 8
    const f4 m = *(const f4*)(choice + b * MUX_IN + (tid & 7) * 4);

    const float* src = in + base_f4 * 4;   // uniform -> SGPR pair (saddr)
    f4* dst = (f4*)out + base_f4;

    // ---- prologue: fill pipeline with DEPTH-1 async loads --------------
#pragma unroll
    for (int k = 0; k < DEPTH - 1; ++k) {
        const unsigned lds = (unsigned)(size_t)&lbuf[k][tid]; // low 32b = LDS offset
        const unsigned go  = (unsigned)((k * TPB + tid) * sizeof(f4));
        async_load_b128_nt(lds, go, src);
    }

    // ---- steady state ---------------------------------------------------
    for (int k = 0; k < ITERS; ++k) {
        const int kn = k + DEPTH - 1;
        if (kn < ITERS) {
            // buffer (kn % DEPTH) was consumed at iteration kn-DEPTH; its
            // ds_load data was used (DScnt wait) before this issue in
            // program order -> WAR safe.
            const unsigned lds = (unsigned)(size_t)&lbuf[kn & (DEPTH - 1)][tid];
            const unsigned go  = (unsigned)((kn * TPB + tid) * sizeof(f4));
            async_load_b128_nt(lds, go, src);
        }
        // ensure load k has landed in LDS (async loads signal done in order)
        const int rem = ITERS - 1 - k;
        wait_async(rem < (DEPTH - 1) ? rem : (DEPTH - 1));

        f4 v = lbuf[k & (DEPTH - 1)][tid];   // ds_load_b128
        v *= m;                              // VALU multiply
        __builtin_nontemporal_store(v, dst + (size_t)k * TPB + tid); // b128 th:TH_STORE_NT
    }
}

// ---------------------------------------------------------------------------
extern "C" void kernel_launch(void* const* d_in, const int* in_sizes, int n_in,
                              void* d_out, int out_size, void* d_ws, size_t ws_size,
                              hipStream_t stream) {
    const float* inputs = (const float*)d_in[0];   // (BS, H, W, 32) f32
    const float* u      = (const float*)d_in[1];   // (BS, 1, 32)    f32
    const float* logits = (const float*)d_in[2];   // (1, 32)        f32
    float* out          = (float*)d_out;
    float* choice       = (float*)d_ws;            // BS*32 floats of scratch

    const int BS = in_sizes[1] / MUX_IN;                       // 32
    const long long total_f4 = (long long)in_sizes[0] / 4;     // 16M
    const long long f4_per_batch = total_f4 / BS;              // 512K
    const int chunks = (int)(f4_per_batch / ((long long)TPB * ITERS)); // 64

    choice_kernel<<<BS, MUX_IN, 0, stream>>>(u, logits, choice);
    scale_kernel<<<BS * chunks, TPB, 0, stream>>>(inputs, choice, out, chunks);
}